// OwnRefTransformerBlock_23287312679392
// MI455X (gfx1250) — compile-verified
//
#include <hip/hip_runtime.h>
#include <math.h>

// Problem constants (match reference)
#define B_   512
#define T_   96
#define QD_  8
#define KVD_ 5
#define H_   3
#define DM_  15    // KVD*H
#define DF_  60    // 4*DM
#define NN_  95    // T-1
#define EPS_ 1e-5f

typedef __attribute__((ext_vector_type(2))) float v2f;
typedef __attribute__((ext_vector_type(8))) float v8f;

// Workspace layout (floats):
//   xa  : [0, 49152*16)            LN1 output, 16-float padded rows
//   wf1 : [XA_N, XA_N+1024)        W1 fragments, (kk*4+nt)*32+lane -> 2 floats
//   wf2 : [.., +1024)              W2 fragments, kk*32+lane -> 2 floats
//   b1p : [.., +64)                b1 padded to 64
#define XA_N   (B_ * T_ * 16)
#define WF1_O  XA_N
#define WF2_O  (WF1_O + 1024)
#define B1P_O  (WF2_O + 1024)

// ---------------------------------------------------------------------------
// Kernel 0: pack W1/W2 into WMMA B-fragment order, pad b1. One 512-thread
// block; runs once per launch (deterministic).
// B (4x16) fragment layout: lane L -> N=L&15, vgpr v holds K = v + 2*(L>>4)
// ---------------------------------------------------------------------------
__global__ __launch_bounds__(512) void pack_weights_kernel(
    const float* __restrict__ W1, const float* __restrict__ b1,
    const float* __restrict__ W2,
    float* __restrict__ wf1, float* __restrict__ wf2, float* __restrict__ b1p)
{
  const int i    = threadIdx.x;        // 0..511
  const int lane = i & 31;
  const int half = lane >> 4;
  const int l15  = lane & 15;

  // W1 fragments: frag = kk*4+nt (16 frags x 32 lanes = 512 pairs)
  {
    const int frag = i >> 5;
    const int kk = frag >> 2, nt = frag & 3;
    const int k0 = kk * 4 + half * 2;
    const int n  = nt * 16 + l15;
    wf1[i * 2 + 0] = (k0     < DM_ && n < DF_) ? W1[(k0    ) * DF_ + n] : 0.f;
    wf1[i * 2 + 1] = (k0 + 1 < DM_ && n < DF_) ? W1[(k0 + 1) * DF_ + n] : 0.f;
  }
  // W2 fragments: kk = 0..15 (16 x 32 = 512 pairs)
  {
    const int kk = i >> 5;
    const int k0 = kk * 4 + half * 2;
    wf2[i * 2 + 0] = (k0     < DF_ && l15 < DM_) ? W2[(k0    ) * DM_ + l15] : 0.f;
    wf2[i * 2 + 1] = (k0 + 1 < DF_ && l15 < DM_) ? W2[(k0 + 1) * DM_ + l15] : 0.f;
  }
  if (i < 64) b1p[i] = (i < DF_) ? b1[i] : 0.f;
}

// ---------------------------------------------------------------------------
// Kernel A: fused neighbor-feature gen + attention + LN1.
// One wave32 per token; 3 neighbors per lane. Writes LN1 output as padded
// 16-float rows into workspace (xa).
// ---------------------------------------------------------------------------
__global__ __launch_bounds__(128) void attn_ln1_kernel(
    const float* __restrict__ state, const float* __restrict__ Wq,
    const float* __restrict__ Wk,    const float* __restrict__ Wv,
    const float* __restrict__ g1,    const float* __restrict__ be1,
    float* __restrict__ xa)
{
  const int wave = threadIdx.x >> 5;
  const int lane = threadIdx.x & 31;
  const int tok  = blockIdx.x * 4 + wave;           // 0 .. B*T-1 (exact)
  const int b    = tok / T_;
  const int t    = tok - b * T_;
  const float* srow = state + tok * QD_;

  const float pi0 = srow[0], pi1 = srow[1], pi2 = srow[2], pi3 = srow[3];

  // q = state_row @ Wq (15), computed redundantly per lane (uniform)
  float q[DM_];
#pragma unroll
  for (int d = 0; d < DM_; ++d) {
    float acc = 0.f;
#pragma unroll
    for (int e = 0; e < QD_; ++e) acc += srow[e] * Wq[e * DM_ + d];
    q[d] = acc;
  }
  // p[c][h] = sum_k Wk[c, h*5+k] * q[h*5+k]  (keys are never materialized)
  float p[KVD_][H_];
#pragma unroll
  for (int c = 0; c < KVD_; ++c)
#pragma unroll
    for (int h = 0; h < H_; ++h) {
      float acc = 0.f;
#pragma unroll
      for (int k = 0; k < KVD_; ++k) acc += Wk[c * DM_ + h * KVD_ + k] * q[h * KVD_ + k];
      p[c][h] = acc;
    }

  const float scale   = 0.44721359549995794f; // 1/sqrt(5)
  const float NEG_BIG = -1e30f;

  float kvv[3][KVD_];
  float sc[3][H_];
#pragma unroll
  for (int rep = 0; rep < 3; ++rep) {
    const int  n     = lane + rep * 32;
    const bool valid = (n < NN_);
    int j = (n < t) ? n : n + 1;
    if (!valid) j = t;                         // in-bounds dummy
    const float* sj = state + (b * T_ + j) * QD_;
    const float d0 = sj[0] - pi0, d1 = sj[1] - pi1;
    const float d2 = sj[2] - pi2, d3 = sj[3] - pi3;
    const float ang = 1.57079632679489662f - sj[7];
    const float ca = cosf(ang), sa = sinf(ang);
    // final rel = [d0*ca-d1*sa, d0*sa+d1*ca, d2*ca-d3*sa, d2*sa+d3*ca]
    const float r0 = d0 * ca - d1 * sa;
    const float r1 = d0 * sa + d1 * ca;
    const float r2 = d2 * ca - d3 * sa;
    const float r3 = d2 * sa + d3 * ca;
    const float r    = sqrtf(r0 * r0 + r1 * r1);
    const float invr = 1.0f / r;
    const float rt   = 1.0f / (1.0f + __expf(5.0f * r - 2.0f)); // sigmoid(1-5(r-0.2))
    float k0 = r0 * invr, k1 = r1 * invr, k2 = r2, k3 = r3, k4 = rt;
    if (!valid) { k0 = k1 = k2 = k3 = k4 = 0.f; }
    kvv[rep][0] = k0; kvv[rep][1] = k1; kvv[rep][2] = k2;
    kvv[rep][3] = k3; kvv[rep][4] = k4;
#pragma unroll
    for (int h = 0; h < H_; ++h) {
      float s = k0 * p[0][h] + k1 * p[1][h] + k2 * p[2][h] + k3 * p[3][h] + k4 * p[4][h];
      sc[rep][h] = valid ? s * scale : NEG_BIG;
    }
  }

  // softmax over the 95 neighbors (3 local + wave32 butterfly), then
  // moment matrix m[h][c] = sum_n w[h,n]*kv[n,c]
  float m[H_][KVD_];
#pragma unroll
  for (int h = 0; h < H_; ++h) {
    float mx = fmaxf(fmaxf(sc[0][h], sc[1][h]), sc[2][h]);
#pragma unroll
    for (int off = 16; off > 0; off >>= 1) mx = fmaxf(mx, __shfl_xor(mx, off, 32));
    float e0 = __expf(sc[0][h] - mx);
    float e1 = __expf(sc[1][h] - mx);
    float e2 = __expf(sc[2][h] - mx);
    float s  = e0 + e1 + e2;
#pragma unroll
    for (int off = 16; off > 0; off >>= 1) s += __shfl_xor(s, off, 32);
    const float invs = 1.0f / s;
    const float w0 = e0 * invs, w1 = e1 * invs, w2 = e2 * invs;
#pragma unroll
    for (int c = 0; c < KVD_; ++c)
      m[h][c] = w0 * kvv[0][c] + w1 * kvv[1][c] + w2 * kvv[2][c];
  }
  // cross-lane reduce m (15 values)
#pragma unroll
  for (int h = 0; h < H_; ++h)
#pragma unroll
    for (int c = 0; c < KVD_; ++c) {
      float v = m[h][c];
#pragma unroll
      for (int off = 16; off > 0; off >>= 1) v += __shfl_xor(v, off, 32);
      m[h][c] = v;
    }

  // out[h*5+k] = sum_c m[h][c] * Wv[c, h*5+k]; then LN1
  float o[DM_];
#pragma unroll
  for (int h = 0; h < H_; ++h)
#pragma unroll
    for (int k = 0; k < KVD_; ++k) {
      float acc = 0.f;
#pragma unroll
      for (int c = 0; c < KVD_; ++c) acc += m[h][c] * Wv[c * DM_ + h * KVD_ + k];
      o[h * KVD_ + k] = acc;
    }
  float mu = 0.f;
#pragma unroll
  for (int d = 0; d < DM_; ++d) mu += o[d];
  mu *= (1.0f / DM_);
  float var = 0.f;
#pragma unroll
  for (int d = 0; d < DM_; ++d) { float dd = o[d] - mu; var += dd * dd; }
  var *= (1.0f / DM_);
  const float inv = rsqrtf(var + EPS_);
  if (lane == 0) {
    float* xr = xa + tok * 16;
#pragma unroll
    for (int d = 0; d < DM_; ++d) xr[d] = (o[d] - mu) * inv * g1[d] + be1[d];
    xr[15] = 0.f;                              // K-padding for WMMA
  }
}

// ---------------------------------------------------------------------------
// Kernel B: MLP (15->60 relu ->15) via V_WMMA_F32_16X16X4_F32, + residual +
// LN2. One wave per 16-token tile; hidden activation re-fragmented via LDS.
// Fragment layouts per ISA 7.12.2:
//   A (16x4): lane L -> M=L&15, vgpr v holds K = v + 2*(L>>4)
//   B (4x16): lane L -> N=L&15, vgpr v holds K = v + 2*(L>>4)  (pre-packed)
//   C/D     : lane L -> N=L&15, vgpr v holds M = v + 8*(L>>4)
// ---------------------------------------------------------------------------
#define WAVES_B 4
#define H1S 66   // even -> 8B-aligned v2f LDS reads at even k0
__global__ __launch_bounds__(128) void mlp_ln2_kernel(
    const float* __restrict__ xa,  const float* __restrict__ wf1,
    const float* __restrict__ b1p, const float* __restrict__ wf2,
    const float* __restrict__ b2,  const float* __restrict__ g2,
    const float* __restrict__ be2, float* __restrict__ out)
{
  __shared__ float h1s[WAVES_B][16 * H1S];  // hidden, padded stride 66
  __shared__ float ys [WAVES_B][16 * 17];   // gemm2 result, padded stride 17

  const int wave = threadIdx.x >> 5;
  const int lane = threadIdx.x & 31;
  const int half = lane >> 4;
  const int l15  = lane & 15;
  const int tile = blockIdx.x * WAVES_B + wave;   // 0..3071 (exact)
  const int row0 = tile * 16;

  const v2f* wf1v = (const v2f*)wf1;
  const v2f* wf2v = (const v2f*)wf2;

  // ---- GEMM1: X(16x16, K padded) @ W1p(16x64, N padded) ----
  v8f acc[4] = {};
  const float* xrow = xa + (row0 + l15) * 16;
#pragma unroll
  for (int kk = 0; kk < 4; ++kk) {
    const int k0 = kk * 4 + half * 2;
    const v2f a = *(const v2f*)(xrow + k0);           // global_load_b64
#pragma unroll
    for (int nt = 0; nt < 4; ++nt) {
      const v2f bb = wf1v[(kk * 4 + nt) * 32 + lane]; // coalesced b64, no guards
      acc[nt] = __builtin_amdgcn_wmma_f32_16x16x4_f32(
          false, a, false, bb, (short)0, acc[nt], false, false);
    }
  }

  // bias + relu, spill to LDS in row-major (m x 64, stride 66)
  float* h1 = h1s[wave];
#pragma unroll
  for (int nt = 0; nt < 4; ++nt) {
    const int n = nt * 16 + l15;
    const float bias = b1p[n];                        // padded, no guard
#pragma unroll
    for (int v = 0; v < 8; ++v) {
      const int mrow = v + half * 8;
      h1[mrow * H1S + n] = fmaxf(acc[nt][v] + bias, 0.f);  // n>=60 stays 0
    }
  }
  __syncthreads();

  // ---- GEMM2: H1(16x64) @ W2p(64x16) ----
  v8f acc2 = {};
#pragma unroll
  for (int kk = 0; kk < 16; ++kk) {
    const int k0 = kk * 4 + half * 2;
    const v2f a = *(const v2f*)(&h1[l15 * H1S + k0]); // ds_load_b64 (aligned)
    const v2f bb = wf2v[kk * 32 + lane];              // coalesced b64, no guards
    acc2 = __builtin_amdgcn_wmma_f32_16x16x4_f32(
        false, a, false, bb, (short)0, acc2, false, false);
  }

  float* y = ys[wave];
#pragma unroll
  for (int v = 0; v < 8; ++v) y[(v + half * 8) * 17 + l15] = acc2[v];
  __syncthreads();

  // ---- bias + residual + LN2, one row per lane (lanes 0-15) ----
  if (half == 0) {
    const int row = row0 + l15;
    const float* xr = xa + row * 16;
    float vals[DM_];
    float mu = 0.f;
#pragma unroll
    for (int n = 0; n < DM_; ++n) {
      const float v = y[l15 * 17 + n] + b2[n] + xr[n];
      vals[n] = v;
      mu += v;
    }
    mu *= (1.0f / DM_);
    float var = 0.f;
#pragma unroll
    for (int n = 0; n < DM_; ++n) { const float d = vals[n] - mu; var += d * d; }
    var *= (1.0f / DM_);
    const float inv = rsqrtf(var + EPS_);
    float* orow = out + row * DM_;
#pragma unroll
    for (int n = 0; n < DM_; ++n)
      orow[n] = (vals[n] - mu) * inv * g2[n] + be2[n];
  }
}

// ---------------------------------------------------------------------------
extern "C" void kernel_launch(void* const* d_in, const int* in_sizes, int n_in,
                              void* d_out, int out_size, void* d_ws, size_t ws_size,
                              hipStream_t stream) {
  const float* state = (const float*)d_in[0];
  const float* Wq    = (const float*)d_in[1];
  const float* Wk    = (const float*)d_in[2];
  const float* Wv    = (const float*)d_in[3];
  const float* ln1_g = (const float*)d_in[4];
  const float* ln1_b = (const float*)d_in[5];
  const float* W1    = (const float*)d_in[6];
  const float* b1    = (const float*)d_in[7];
  const float* W2    = (const float*)d_in[8];
  const float* b2    = (const float*)d_in[9];
  const float* ln2_g = (const float*)d_in[10];
  const float* ln2_b = (const float*)d_in[11];
  float* outp = (float*)d_out;

  float* ws  = (float*)d_ws;
  float* xa  = ws;            // (B*T) x 16 padded LN1 output
  float* wf1 = ws + WF1_O;    // packed W1 fragments
  float* wf2 = ws + WF2_O;    // packed W2 fragments
  float* b1p = ws + B1P_O;    // padded b1

  const int tokens = B_ * T_;                  // 49152
  pack_weights_kernel<<<1, 512, 0, stream>>>(W1, b1, W2, wf1, wf2, b1p);
  attn_ln1_kernel<<<tokens / 4, 128, 0, stream>>>(state, Wq, Wk, Wv, ln1_g, ln1_b, xa);
  mlp_ln2_kernel<<<(tokens / 16) / WAVES_B, 128, 0, stream>>>(xa, wf1, b1p, wf2,
                                                              b2, ln2_g, ln2_b, outp);
}